// OccamNet_24507083391427
// MI455X (gfx1250) — compile-verified
//
#include <hip/hip_runtime.h>
#include <math.h>

typedef __attribute__((ext_vector_type(2))) float v2f;
typedef __attribute__((ext_vector_type(8))) float v8f;

#define B_TOTAL   2097152
#define NTILES    (B_TOTAL / 16)
#define NLAYER    5
#define LDS_STRIDE 28   // max feature width

// broadcast value from lane `src` (0..15) to all lanes of the wave32
__device__ __forceinline__ float bcast16(float v, int src) {
    int r = __builtin_amdgcn_ds_bpermute(src << 2, __float_as_int(v));
    return __int_as_float(r);
}

// One layer: acc(16x16 f32, args^T) = sum_c A_chunk(16x4) * B_chunk(4x16).
// bv[c] already holds exactly this lane's B half-pair -> no per-chunk VALU.
#define DO_LAYER(C, ABASE)                                                     \
    {                                                                          \
        acc = (v8f){0.f, 0.f, 0.f, 0.f, 0.f, 0.f, 0.f, 0.f};                   \
        _Pragma("unroll") for (int c = 0; c < (C); ++c) {                      \
            acc = __builtin_amdgcn_wmma_f32_16x16x4_f32(                       \
                false, Areg[(ABASE) + c], false, bv[c], (short)0, acc,         \
                false, false);                                                 \
        }                                                                      \
    }

// Commit feature f (broadcast from lane oo) into this lane's B-layout slot.
// Feature f lives in chunk f>>2, lane-half (f>>1)&1, element f&1 — all
// compile-time; the only runtime op is one v_cndmask on `keep`.
#define PUT_FEAT(f, val)                                                       \
    {                                                                          \
        float nv = bcast16((val), oo);                                         \
        const bool keep = (hi == (((f) >> 1) & 1));                            \
        if (((f) & 1))                                                         \
            bv[(f) >> 2].y = keep ? nv : bv[(f) >> 2].y;                       \
        else                                                                   \
            bv[(f) >> 2].x = keep ? nv : bv[(f) >> 2].x;                       \
    }

// Bases on the 8 args: valid in lanes 0..15 (lane n holds args of batch
// column n in acc[0..7]); append as features W..W+5.
#define DO_IMG(W)                                                              \
    {                                                                          \
        float i0 = acc[0] + acc[1];                                            \
        float i1 = acc[2] * acc[3];                                            \
        float i2 = __sinf(acc[4]);                                             \
        float i3 = __cosf(acc[5]);                                             \
        float i4 = acc[6] * acc[6];                                            \
        float i5 = acc[7];                                                     \
        PUT_FEAT((W) + 0, i0)                                                  \
        PUT_FEAT((W) + 1, i1)                                                  \
        PUT_FEAT((W) + 2, i2)                                                  \
        PUT_FEAT((W) + 3, i3)                                                  \
        PUT_FEAT((W) + 4, i4)                                                  \
        PUT_FEAT((W) + 5, i5)                                                  \
    }

__global__ void __launch_bounds__(256)
occamnet_wmma(const float* __restrict__ x,
              const float* __restrict__ wsrc, const float* __restrict__ wh0,
              const float* __restrict__ wh1,  const float* __restrict__ wh2,
              const float* __restrict__ wlast, float* __restrict__ out)
{
    // Softmaxed weights, column-permuted into "append" feature order:
    // features = [x0,x1,x2,1, img0(6), img1(6), img2(6), img3(6)].
    // Rows 8..15 (and rows 1..7 of the last layer) stay zero -> padded A.
    __shared__ float smw[NLAYER][16][LDS_STRIDE];

    const int tid = threadIdx.x;
    for (int i = tid; i < NLAYER * 16 * LDS_STRIDE; i += blockDim.x)
        ((float*)smw)[i] = 0.0f;
    __syncthreads();

    // 33 softmax rows: layers 0..3 have 8 rows, layer 4 has 1.
    if (tid < 33) {
        int l, o;
        if (tid < 32) { l = tid >> 3; o = tid & 7; } else { l = 4; o = 0; }
        const float* w = (l == 0) ? wsrc : (l == 1) ? wh0 : (l == 2) ? wh1
                        : (l == 3) ? wh2 : wlast;
        const int W = 4 + 6 * l;                 // in-features of this layer
        const float* row = w + o * W;            // torch Linear: [out, in]
        float m = row[0];
        for (int i = 1; i < W; ++i) m = fmaxf(m, row[i]);
        float s = 0.0f;
        for (int i = 0; i < W; ++i) s += __expf(row[i] - m);
        const float inv = 1.0f / s;
        for (int r = 0; r < W; ++r) {            // r = reference column
            float val = __expf(row[r] - m) * inv;
            int f;                               // my (appended) feature index
            if (r >= W - 4) {                    // x block (reference: last)
                f = r - (W - 4);
            } else {                             // img blocks, newest first
                int jb = r / 6, k = r - 6 * jb;
                f = 4 + 6 * ((l - 1) - jb) + k;
            }
            smw[l][o][f] = val;
        }
    }
    __syncthreads();

    const int lane = threadIdx.x & 31;
    const int oo   = lane & 15;        // output row (M) / batch column (N)
    const int hi   = lane >> 4;        // which K half-pair this lane supplies

    // Preload loop-invariant A chunks (16x4 f32 A-layout:
    // lane L holds M=L&15, VGPR0 -> K=2*(L>=16), VGPR1 -> K=2*(L>=16)+1).
    v2f Areg[21];
#define LOADA(l, base, C)                                                      \
    _Pragma("unroll") for (int c = 0; c < (C); ++c) {                          \
        int k0 = 4 * c + 2 * hi;                                               \
        Areg[(base) + c] = (v2f){smw[l][oo][k0], smw[l][oo][k0 + 1]};          \
    }
    LOADA(0, 0, 1)   // W=4  -> 1 chunk
    LOADA(1, 1, 3)   // W=10 -> 3 chunks (padded to 12)
    LOADA(2, 4, 4)   // W=16 -> 4 chunks
    LOADA(3, 8, 6)   // W=22 -> 6 chunks (padded to 24)
    LOADA(4, 14, 7)  // W=28 -> 7 chunks
#undef LOADA

    const int wavesPerBlk = blockDim.x >> 5;
    const int waveId = blockIdx.x * wavesPerBlk + (threadIdx.x >> 5);
    const int nWaves = gridDim.x * wavesPerBlk;

    v8f acc;
    v2f bv[7];      // this lane's B-layout feature slots (K = 4c+2*hi+{0,1})
    for (int tile = waveId; tile < NTILES; tile += nWaves) {
        const int col = tile * 16 + oo;   // batch column handled by this lane
        const float* xr = x + col * 3;
        bv[0].x = xr[2 * hi];             // hi ? x2 : x0  (address math only)
        float x1v = xr[1];
        bv[0].y = hi ? 1.0f : x1v;        // hi half holds the constant column
#pragma unroll
        for (int c = 1; c < 7; ++c) bv[c] = (v2f){0.f, 0.f};  // zero padding

        DO_LAYER(1, 0)   DO_IMG(4)    // source: [B,4]  -> args
        DO_LAYER(3, 1)   DO_IMG(10)   // h0:     [B,10] -> args
        DO_LAYER(4, 4)   DO_IMG(16)   // h1:     [B,16] -> args
        DO_LAYER(6, 8)   DO_IMG(22)   // h2:     [B,22] -> args
        DO_LAYER(7, 14)               // last:   [B,28] -> 1 output (row M=0)

        if (hi == 0)                  // lanes 0..15 hold out[col] in acc[0]
            out[tile * 16 + oo] = acc[0];
    }
}

extern "C" void kernel_launch(void* const* d_in, const int* in_sizes, int n_in,
                              void* d_out, int out_size, void* d_ws, size_t ws_size,
                              hipStream_t stream) {
    const float* x     = (const float*)d_in[0];
    const float* wsrc  = (const float*)d_in[1];
    const float* wh0   = (const float*)d_in[2];
    const float* wh1   = (const float*)d_in[3];
    const float* wh2   = (const float*)d_in[4];
    const float* wlast = (const float*)d_in[5];
    float* out = (float*)d_out;

    dim3 grid(1024), block(256);   // 8192 waves, 16 tiles each (grid-stride)
    occamnet_wmma<<<grid, block, 0, stream>>>(x, wsrc, wh0, wh1, wh2, wlast, out);
}